// ChamferLoss_loss_26259430047859
// MI455X (gfx1250) — compile-verified
//
#include <hip/hip_runtime.h>
#include <math.h>

// Chamfer loss via V_WMMA_F32_16X16X4_F32:
//   A row n: [-2px, -2py, -2pz, 1]   B col m: [gx, gy, gz, g2]   C = 0
//   D = -2 p.g + |g|^2 ;  min_m d2 = |p|^2 + min_m D   (p2 added post-reduction)
// sqrt/max(.,EPS) deferred to the epilogue (monotone, commutes with min).

typedef float v2f __attribute__((ext_vector_type(2)));
typedef float v8f __attribute__((ext_vector_type(8)));

#define TILE 16
#define ROWS_PER_WAVE 2
#define WAVES_PER_BLOCK 4
#define ROWS_PER_BLOCK (TILE * ROWS_PER_WAVE * WAVES_PER_BLOCK)  // 128

// Raw v_min_num_f32: skips the compiler's canonicalize (v_max x,x) insertion.
// Safe here: operands are finite arithmetic results, never sNaN.
__device__ __forceinline__ float fmin_raw(float a, float b) {
    float r;
    asm("v_min_num_f32 %0, %1, %2" : "=v"(r) : "v"(a), "v"(b));
    return r;
}

__global__ __launch_bounds__(WAVES_PER_BLOCK * 32)
void chamfer_min_dir(const float* __restrict__ P, const float* __restrict__ G,
                     float* __restrict__ outMin, int Nrow, int Ncol) {
    const int lane = threadIdx.x & 31;
    const int wave = threadIdx.x >> 5;
    const int rowBlocks = Nrow / ROWS_PER_BLOCK;
    const int b  = blockIdx.x / rowBlocks;
    const int rb = blockIdx.x % rowBlocks;
    const long long pOff = (long long)b * 3 * Nrow;
    const long long gOff = (long long)b * 3 * Ncol;
    const int  l16 = lane & 15;
    const bool lo  = (lane < 16);

    // ---- A strips (predict points): [-2px, -2py, -2pz, 1] ------------------
    // A layout (16x4 f32): lanes 0-15 hold K=0,1; lanes 16-31 hold K=2,3; M=lane%16.
    v2f a[ROWS_PER_WAVE];
    v8f acc[ROWS_PER_WAVE];
    float p2s[ROWS_PER_WAVE];
    int rowBase[ROWS_PER_WAVE];
#pragma unroll
    for (int r = 0; r < ROWS_PER_WAVE; ++r) {
        rowBase[r] = rb * ROWS_PER_BLOCK + wave * (ROWS_PER_WAVE * TILE) + r * TILE;
        const int n = rowBase[r] + l16;
        const float px = P[pOff + n];
        const float py = P[pOff + Nrow + n];
        const float pz = P[pOff + 2 * Nrow + n];
        p2s[r] = px * px + py * py + pz * pz;
        a[r].x = lo ? -2.0f * px : -2.0f * pz;   // K0 | K2
        a[r].y = lo ? -2.0f * py : 1.0f;         // K1 | K3
#pragma unroll
        for (int j = 0; j < 8; ++j) acc[r][j] = INFINITY;
    }

    const v8f czero = { 0.f, 0.f, 0.f, 0.f, 0.f, 0.f, 0.f, 0.f };  // loop-invariant C

    // ---- Sweep all gt tiles ------------------------------------------------
    // B layout (4x16 f32): lanes 0-15 hold K=0,1; lanes 16-31 hold K=2,3; N=lane%16.
    const int tiles = Ncol / TILE;
    for (int t = 0; t < tiles; ++t) {
        const int m = t * TILE + l16;
        const float gx = G[gOff + m];
        const float gy = G[gOff + Ncol + m];
        const float gz = G[gOff + 2 * Ncol + m];
        const float g2 = gx * gx + gy * gy + gz * gz;
        v2f bb;
        bb.x = lo ? gx : gz;                     // K0 | K2
        bb.y = lo ? gy : g2;                     // K1 | K3

#pragma unroll
        for (int r = 0; r < ROWS_PER_WAVE; ++r) {
            // (neg_a, A, neg_b, B, c_mod, C, reuse_a, reuse_b)
            v8f d = __builtin_amdgcn_wmma_f32_16x16x4_f32(
                false, a[r], false, bb, (short)0, czero, false, false);
#pragma unroll
            for (int j = 0; j < 8; ++j) acc[r][j] = fmin_raw(acc[r][j], d[j]);
        }
    }

    // ---- Min across the 16 columns (lanes) of each half --------------------
    // D layout: lanes 0-15 -> M = j, lanes 16-31 -> M = j+8; N = lane%16.
    // xor masks 1,2,4,8 stay inside each 16-lane half of the wave32.
#pragma unroll
    for (int r = 0; r < ROWS_PER_WAVE; ++r)
#pragma unroll
        for (int j = 0; j < 8; ++j)
#pragma unroll
            for (int off = 1; off < 16; off <<= 1)
                acc[r][j] = fmin_raw(acc[r][j], __shfl_xor(acc[r][j], off, 32));

    // Lane (half,l16=j) stores row j+8*half; fetch that row's |p|^2 (held by
    // lane l16+8*half in the lower half) with one shuffle per strip.
    const int half = lane >> 4;
#pragma unroll
    for (int r = 0; r < ROWS_PER_WAVE; ++r) {
        const float p2w = __shfl(p2s[r], l16 + 8 * half, 32);
#pragma unroll
        for (int j = 0; j < 8; ++j) {
            if (l16 == j)
                outMin[(long long)b * Nrow + rowBase[r] + half * 8 + j] =
                    acc[r][j] + p2w;
        }
    }
}

// ---- Epilogue: loss = sum(sqrt(max(min_d2, EPS))) / denom ------------------
__global__ __launch_bounds__(256)
void chamfer_reduce(const float* __restrict__ mins, int total, float invDenom,
                    float* __restrict__ out) {
    __shared__ float sm[256];
    float s = 0.0f;
    for (int i = threadIdx.x; i < total; i += 256)
        s += sqrtf(fmaxf(mins[i], 1e-12f));
    sm[threadIdx.x] = s;
    __syncthreads();
#pragma unroll
    for (int off = 128; off > 0; off >>= 1) {
        if (threadIdx.x < off) sm[threadIdx.x] += sm[threadIdx.x + off];
        __syncthreads();
    }
    if (threadIdx.x == 0) out[0] = sm[0] * invDenom;
}

extern "C" void kernel_launch(void* const* d_in, const int* in_sizes, int n_in,
                              void* d_out, int out_size, void* d_ws, size_t ws_size,
                              hipStream_t stream) {
    const float* predict = (const float*)d_in[0];  // [B, 3, Np]
    const float* gt      = (const float*)d_in[1];  // [B, 3, Ng]
    const int B = 4, C = 3;
    const int Np = in_sizes[0] / (B * C);
    const int Ng = in_sizes[1] / (B * C);

    float* minPred = (float*)d_ws;                    // B * Np floats
    float* minGt   = minPred + (size_t)B * Np;        // B * Ng floats (contiguous)

    const dim3 blk(WAVES_PER_BLOCK * 32);
    // Pass 1: for each predict point, min d2 over all gt points
    chamfer_min_dir<<<B * (Np / ROWS_PER_BLOCK), blk, 0, stream>>>(
        predict, gt, minPred, Np, Ng);
    // Pass 2: roles swapped — for each gt point, min d2 over all predict points
    chamfer_min_dir<<<B * (Ng / ROWS_PER_BLOCK), blk, 0, stream>>>(
        gt, predict, minGt, Ng, Np);

    const int total = B * (Np + Ng);
    const float invDenom = 1.0f / (float)(B * (Ng + Np));
    chamfer_reduce<<<1, 256, 0, stream>>>(minPred, total, invDenom, (float*)d_out);
}